// GHMLoss_39788577030436
// MI455X (gfx1250) — compile-verified
//
#include <hip/hip_runtime.h>
#include <cstdint>

// GHM-C loss, single streaming pass with a CDNA5 async-to-LDS pipeline.
//
// Folded reference math: loss = (1/n) * sum_b clip(w_b/num_nonempty, 1e-4) * S_b
// where count_b / S_b are the per-bin histogram count and per-bin sum of
// BCE-with-logits. One pass accumulates the 20 scalars; a tiny finalize kernel
// applies the EMA / weight / clip logic on 10 bins.
//
// Roofline: 268 MB read once -> ~11.5 us at 23.3 TB/s; compute (~2 trans + ~20
// VALU + 2 ds_add per element) is below that. Data is streamed with
// GLOBAL_LOAD_ASYNC_TO_LDS_B128 (double-buffered, per-thread-owned slots, so
// only s_wait_asynccnt is needed -- no barriers), keeping zero VGPRs tied to
// outstanding loads. Histogram uses a transposed LDS layout (slot = k*256+tid;
// 256 % 64banks == 0) so every ds_add_f32 is bank-conflict-free regardless of
// the data-dependent bin index.

#define GHM_BINS   10
#define GHM_ALPHA  0.75f
#define GHM_TPB    256
#define GHM_BLOCKS 4096

typedef float v4f __attribute__((ext_vector_type(4)));

// Low 32 bits of a generic pointer to __shared__ = wave-relative LDS byte
// offset (FLAT aperture mapping: the LDS offset is the low address word).
__device__ __forceinline__ unsigned lds_off_u32(const void* p) {
    return (unsigned)(uintptr_t)p;
}

// GVS-mode async copy: copies 16 bytes from global address sbase + voff_bytes to the LDS address in lds_addr. Tracked by ASYNCcnt.
__device__ __forceinline__ void async_load_f4(unsigned lds_addr,
                                              const float* __restrict__ sbase,
                                              unsigned voff_bytes) {
    asm volatile("global_load_async_to_lds_b128 %0, %1, %2 offset:0"
                 :
                 : "v"(lds_addr), "v"(voff_bytes), "s"(sbase)
                 : "memory");
}

__global__ __launch_bounds__(64) void ghm_zero_ws(float* __restrict__ ws) {
    int t = threadIdx.x;
    if (t < 2 * GHM_BINS) ws[t] = 0.0f;
}

__global__ __launch_bounds__(GHM_TPB) void ghm_hist_kernel(
    const float* __restrict__ x,
    const float* __restrict__ tgt,
    float* __restrict__ ws,
    long long n4)
{
    // h[k*TPB + tid]: k in [0,10) = counts, k in [10,20) = bce sums. 20 KB.
    __shared__ float h[2 * GHM_BINS * GHM_TPB];
    __shared__ v4f xstage[2][GHM_TPB];   // 8 KB double-buffered x tiles
    __shared__ v4f tstage[2][GHM_TPB];   // 8 KB double-buffered target tiles
    const int tid = threadIdx.x;

    #pragma unroll
    for (int k = 0; k < 2 * GHM_BINS; ++k)
        h[k * GHM_TPB + tid] = 0.0f;
    __syncthreads();

    const long long stride = (long long)gridDim.x * (long long)blockDim.x;
    long long i = (long long)blockIdx.x * blockDim.x + tid;

    const unsigned xs[2] = { lds_off_u32(&xstage[0][tid]),
                             lds_off_u32(&xstage[1][tid]) };
    const unsigned ts[2] = { lds_off_u32(&tstage[0][tid]),
                             lds_off_u32(&tstage[1][tid]) };

    // Prime the pipeline with tile 0.
    if (i < n4) {
        unsigned off = (unsigned)(i * 16);
        async_load_f4(xs[0], x, off);
        async_load_f4(ts[0], tgt, off);
    }

    int buf = 0;
    for (; i < n4; i += stride) {
        long long nxt = i + stride;
        if (nxt < n4) {
            // Issue next tile, then wait until only its 2 async ops remain
            // (async loads retire in order -> current tile has landed).
            unsigned off = (unsigned)(nxt * 16);
            async_load_f4(xs[buf ^ 1], x, off);
            async_load_f4(ts[buf ^ 1], tgt, off);
            asm volatile("s_wait_asynccnt 2" ::: "memory");
        } else {
            asm volatile("s_wait_asynccnt 0" ::: "memory");
        }

        // ds_load_b128 of this thread's own staged slot (conflict-free).
        v4f xv = xstage[buf][tid];
        v4f tv = tstage[buf][tid];

        #pragma unroll
        for (int c = 0; c < 4; ++c) {
            float xx = xv[c];
            float tt = tv[c];
            float ax = __builtin_fabsf(xx);
            float e  = __expf(-ax);                 // exp(-|x|) in (0,1]
            float inv = 1.0f / (1.0f + e);          // sigmoid(|x|)
            float s  = (xx >= 0.0f) ? inv : (1.0f - inv);  // sigmoid(x)
            float g  = __builtin_fabsf(s - tt);     // grad norm, in [0,1)

            int b = (int)(g * ((float)GHM_BINS - 0.0001f)); // g>=0: trunc==floor
            b = (b < 0) ? 0 : ((b > GHM_BINS - 1) ? GHM_BINS - 1 : b);

            // Stable BCE-with-logits: max(x,0) - x*t + log(1 + exp(-|x|))
            float bce = __builtin_fmaxf(xx, 0.0f) - xx * tt + __logf(1.0f + e);

            // Transposed layout: bank = (b*256 + tid) % 64 = tid % 64.
            atomicAdd(&h[b * GHM_TPB + tid], 1.0f);
            atomicAdd(&h[(GHM_BINS + b) * GHM_TPB + tid], bce);
        }
        buf ^= 1;
    }
    __syncthreads();

    // Tree-reduce the 20 arrays over the tid dimension.
    for (int off = GHM_TPB / 2; off > 0; off >>= 1) {
        if (tid < off) {
            #pragma unroll
            for (int k = 0; k < 2 * GHM_BINS; ++k)
                h[k * GHM_TPB + tid] += h[k * GHM_TPB + tid + off];
        }
        __syncthreads();
    }

    if (tid < 2 * GHM_BINS)
        atomicAdd(&ws[tid], h[tid * GHM_TPB]);
}

__global__ __launch_bounds__(64) void ghm_finalize(
    const float* __restrict__ ws,       // [0..9] counts, [10..19] bce sums
    const float* __restrict__ acc_sum,  // [10]
    float* __restrict__ out,
    float n_total)
{
    if (threadIdx.x == 0 && blockIdx.x == 0) {
        float w[GHM_BINS];
        float sums[GHM_BINS];
        int nn = 0;
        #pragma unroll
        for (int b = 0; b < GHM_BINS; ++b) {
            float cnt = ws[b];
            sums[b] = ws[GHM_BINS + b];
            bool ne = (cnt >= 1.0f);
            float na = ne ? (GHM_ALPHA * acc_sum[b] + (1.0f - GHM_ALPHA) * cnt)
                          : acc_sum[b];
            w[b] = ne ? (n_total / na) : 0.0f;
            nn += ne ? 1 : 0;
        }
        float inv_nn = (nn > 0) ? (1.0f / (float)nn) : 1.0f;
        float loss = 0.0f;
        #pragma unroll
        for (int b = 0; b < GHM_BINS; ++b) {
            float wb = __builtin_fmaxf(w[b] * inv_nn, 0.0001f);
            // Empty bins contribute sums[b] == 0, matching the reference.
            loss += wb * sums[b];
        }
        out[0] = loss / n_total;
    }
}

extern "C" void kernel_launch(void* const* d_in, const int* in_sizes, int n_in,
                              void* d_out, int out_size, void* d_ws, size_t ws_size,
                              hipStream_t stream) {
    const float* x   = (const float*)d_in[0];
    const float* tgt = (const float*)d_in[1];
    const float* acc = (const float*)d_in[2];
    float* ws  = (float*)d_ws;
    float* out = (float*)d_out;

    long long n  = (long long)in_sizes[0];   // 262144 * 128 = 33554432
    long long n4 = n >> 2;                   // float4 count

    ghm_zero_ws<<<1, 64, 0, stream>>>(ws);
    ghm_hist_kernel<<<GHM_BLOCKS, GHM_TPB, 0, stream>>>(x, tgt, ws, n4);
    ghm_finalize<<<1, 64, 0, stream>>>(ws, acc, out, (float)n);
}